// PhaserModel_87316685128386
// MI455X (gfx1250) — compile-verified
//
#include <hip/hip_runtime.h>
#include <hip/hip_bf16.h>

// ---------------------------------------------------------------------------
// Phaser grey-box model for MI455X (gfx1250, wave32).
//   Stage 1 (parallel, WMMA):  p[c][n] = tanh((1-tan(d))/(1+tan(d))),
//       d = bias + depth*0.5*(1 + MLP(cos LFO)); MLP layer 1 via
//       V_WMMA_F32_16X16X4_F32 (16 samples x 16 hidden per instruction),
//       layer 2 via wave32 shfl_xor butterfly. Branchless hw transcendentals.
//   Stage 2 (chunked parallel scan of the linear time-varying 8-state IIR):
//       A) per-chunk 8x9 affine transfer composition (2048 parallel chunks,
//          float4 loads + global_prefetch_b8)
//       B) serial scan over 1024 chunk matrices per channel (cheap)
//       C) parallel replay of the true recurrence + B128 output stores
// ---------------------------------------------------------------------------

#define SR_F       44100.0f
#define MIN_RATE_F 0.1f
#define MAX_RATE_F 5.0f
#define N_SAMP     524288
#define CHUNK      512
#define NCHUNKS    (N_SAMP / CHUNK)   // 1024 per channel
#define TWO_PI_F   6.28318530717958647692f

typedef __attribute__((ext_vector_type(2))) float v2f;
typedef __attribute__((ext_vector_type(8))) float v8f;

// Branchless fast tanh: 1 - 2/(exp(2x)+1).  exp->v_exp_f32, rcp->v_rcp_f32.
// Correct saturation at +/-inf; adequate precision for the grey-box model.
__device__ __forceinline__ float fast_tanh(float x)
{
    const float e = __expf(2.0f * x);
    const float r = __builtin_amdgcn_rcpf(e + 1.0f);
    return fmaf(-2.0f, r, 1.0f);
}

// -------------------------- Stage 1: coefficients --------------------------
__global__ __launch_bounds__(256)
void phaser_coeff_wmma(const float* __restrict__ rate01,
                       const float* __restrict__ phoff01,
                       const float* __restrict__ W1,
                       const float* __restrict__ b1,
                       const float* __restrict__ W2,
                       const float* __restrict__ b2,
                       const float* __restrict__ lfo_amp,
                       const float* __restrict__ bias,
                       const float* __restrict__ depth,
                       const float* __restrict__ prev_phase,
                       float* __restrict__ p_out,
                       int n_per_ch)
{
    const float rate  = rate01[0] * (MAX_RATE_F - MIN_RATE_F) + MIN_RATE_F;
    const float w     = TWO_PI_F * rate / SR_F;
    const float ph0   = prev_phase[0];
    const float phoff = phoff01[0] * TWO_PI_F;
    const float amp   = lfo_amp[0];
    const float bz    = bias[0];
    const float dp    = depth[0];
    const float b2v   = b2[0];

    const int lane = threadIdx.x & 31;
    const int hl   = lane & 15;          // position within 16-lane half
    const bool lo  = lane < 16;          // lower half holds K=0 slices

    // Tile-invariant WMMA operands.
    // B (4x16, K-major): row K=0 holds W1, rows 1..3 are zero padding.
    v2f bmat;
    bmat.x = lo ? W1[hl] : 0.0f;         // VGPR0: lanes0-15 -> K=0 row
    bmat.y = 0.0f;                       // K=1 / K=3 rows -> zero
    // C (16x16): broadcast b1 along rows (C[m][n] = b1[n], n = lane%16).
    const float b1l = b1[hl];
    v8f cmat;
#pragma unroll
    for (int r = 0; r < 8; ++r) cmat[r] = b1l;
    const float w2l = W2[hl];

    const int wave   = (blockIdx.x * blockDim.x + threadIdx.x) >> 5;
    const int nwaves = (gridDim.x * blockDim.x) >> 5;
    const int ntiles = (2 * n_per_ch) >> 4;   // 16 samples per tile

    for (int t = wave; t < ntiles; t += nwaves) {
        const int s0 = t << 4;                      // flat sample base
        const int ch = (s0 >= n_per_ch) ? 1 : 0;
        const int i0 = s0 - ch * n_per_ch;          // index within channel

        // A (16x4): column K=0 holds 16 LFO samples (row M = lane%16).
        // arg <= ~373 rad = ~59 revolutions -> safe for hw v_cos range.
        const float arg = w * (float)(i0 + hl + 1) + ph0 + (ch ? phoff : 0.0f);
        v2f amat;
        amat.x = lo ? (amp * __cosf(arg)) : 0.0f;   // lanes16-31 carry K=2 -> 0
        amat.y = 0.0f;                              // K=1 / K=3 -> 0

        // D = A*B + C  ==  lfo_m * W1[n] + b1[n]   (16 samples x 16 hidden)
        v8f acc = __builtin_amdgcn_wmma_f32_16x16x4_f32(
            false, amat, false, bmat, (short)0, cmat, false, false);

        // Layer 2: m_M = sum_n tanh(acc)[M][n]*W2[n] -> butterfly over 16 lanes
        float tr[8];
#pragma unroll
        for (int r = 0; r < 8; ++r) tr[r] = fast_tanh(acc[r]) * w2l;
#pragma unroll
        for (int r = 0; r < 8; ++r) {
            float v = tr[r];
            v += __shfl_xor(v, 1, 32);
            v += __shfl_xor(v, 2, 32);
            v += __shfl_xor(v, 4, 32);
            v += __shfl_xor(v, 8, 32);
            tr[r] = v;
        }

        // One lane per half finishes the transcendental tail and stores 8 p's.
        if (hl == 0) {
            float pv[8];
#pragma unroll
            for (int r = 0; r < 8; ++r) {
                const float m  = tr[r] + b2v;
                const float d  = bz + dp * 0.5f * (1.0f + m);
                const float td = __tanf(d);
                pv[r] = fast_tanh((1.0f - td) / (1.0f + td));
            }
            float4* dst = (float4*)(p_out + s0 + (lo ? 0 : 8));
            dst[0] = make_float4(pv[0], pv[1], pv[2], pv[3]);
            dst[1] = make_float4(pv[4], pv[5], pv[6], pv[7]);
        }
        // divergence reconverges here; next WMMA executes with full EXEC
    }
}

// ---------------- Stage 2: linear time-varying state recurrence ------------
// State s = (x0,x1,x2,x3, y0,y1,y2,y3); one step of the K=4 cascade:
__device__ __forceinline__ void step_state(float s[8], float xn, float p, float g1)
{
    const float u0 = xn + g1 * s[7];
    const float y0 = fmaf(p, u0, s[0]) - p * s[4];
    const float y1 = fmaf(p, y0, s[1]) - p * s[5];
    const float y2 = fmaf(p, y1, s[2]) - p * s[6];
    const float y3 = fmaf(p, y2, s[3]) - p * s[7];
    s[0] = u0; s[1] = y0; s[2] = y1; s[3] = y2;
    s[4] = y0; s[5] = y1; s[6] = y2; s[7] = y3;
}

// Pass A: compose each chunk's 512 step maps into one 8x9 affine (M | v).
__global__ __launch_bounds__(256)
void chunk_transfer(const float* __restrict__ x,
                    const float* __restrict__ p,
                    const float* __restrict__ g1p,
                    float* __restrict__ T,
                    int n_per_ch)
{
    const int id = blockIdx.x * blockDim.x + threadIdx.x;
    if (id >= 2 * NCHUNKS) return;
    const int ch   = id / NCHUNKS;
    const int cidx = id - ch * NCHUNKS;
    const float g1 = g1p[0];
    const float* xp = x + ch * n_per_ch + cidx * CHUNK;
    const float* pp = p + ch * n_per_ch + cidx * CHUNK;

    // cols[j], j<8: image of basis vector e_j (linear part); cols[8]: offset v
    float M[9][8];
#pragma unroll
    for (int j = 0; j < 9; ++j)
#pragma unroll
        for (int r = 0; r < 8; ++r) M[j][r] = (j == r) ? 1.0f : 0.0f;

    for (int n = 0; n < CHUNK; n += 4) {
        const float4 xv = *(const float4*)(xp + n);
        const float4 pv = *(const float4*)(pp + n);
        __builtin_prefetch(xp + n + 32, 0, 1);   // -> global_prefetch_b8
        __builtin_prefetch(pp + n + 32, 0, 1);
        const float xs[4] = {xv.x, xv.y, xv.z, xv.w};
        const float ps[4] = {pv.x, pv.y, pv.z, pv.w};
#pragma unroll
        for (int q = 0; q < 4; ++q) {
#pragma unroll
            for (int j = 0; j < 8; ++j) step_state(M[j], 0.0f, ps[q], g1);
            step_state(M[8], xs[q], ps[q], g1);
        }
    }

    float* dst = T + (size_t)id * 72;
#pragma unroll
    for (int j = 0; j < 9; ++j)
#pragma unroll
        for (int r = 0; r < 8; ++r) dst[j * 8 + r] = M[j][r];
}

// Pass B: serial scan over chunk affines -> exclusive-prefix chunk states.
__global__ void chunk_scan(const float* __restrict__ T,
                           float* __restrict__ S)
{
    const int ch = blockIdx.x * blockDim.x + threadIdx.x;
    if (ch >= 2) return;
    float s[8] = {0.f, 0.f, 0.f, 0.f, 0.f, 0.f, 0.f, 0.f};
    for (int c = 0; c < NCHUNKS; ++c) {
        const int idx = ch * NCHUNKS + c;
        float* dst = S + (size_t)idx * 8;
#pragma unroll
        for (int r = 0; r < 8; ++r) dst[r] = s[r];
        const float* t = T + (size_t)idx * 72;
        float ns[8];
#pragma unroll
        for (int r = 0; r < 8; ++r) {
            float acc = t[64 + r];            // offset column v
#pragma unroll
            for (int j = 0; j < 8; ++j) acc = fmaf(t[j * 8 + r], s[j], acc);
            ns[r] = acc;
        }
#pragma unroll
        for (int r = 0; r < 8; ++r) s[r] = ns[r];
    }
}

// Pass C: replay true recurrence per chunk from its known initial state.
__global__ __launch_bounds__(256)
void chunk_emit(const float* __restrict__ x,
                const float* __restrict__ p,
                const float* __restrict__ S,
                const float* __restrict__ g1p,
                float* __restrict__ out,
                int n_per_ch)
{
    const int id = blockIdx.x * blockDim.x + threadIdx.x;
    if (id >= 2 * NCHUNKS) return;
    const int ch   = id / NCHUNKS;
    const int cidx = id - ch * NCHUNKS;
    const float g1 = g1p[0];

    float s[8];
    const float* sp = S + (size_t)id * 8;
#pragma unroll
    for (int r = 0; r < 8; ++r) s[r] = sp[r];

    const float* xp = x + ch * n_per_ch + cidx * CHUNK;
    const float* pp = p + ch * n_per_ch + cidx * CHUNK;
    float* op       = out + ch * n_per_ch + cidx * CHUNK;

    for (int n = 0; n < CHUNK; n += 4) {
        const float4 xv = *(const float4*)(xp + n);
        const float4 pv = *(const float4*)(pp + n);
        __builtin_prefetch(xp + n + 32, 0, 1);   // -> global_prefetch_b8
        __builtin_prefetch(pp + n + 32, 0, 1);
        const float xs[4] = {xv.x, xv.y, xv.z, xv.w};
        const float ps[4] = {pv.x, pv.y, pv.z, pv.w};
        float os[4];
#pragma unroll
        for (int q = 0; q < 4; ++q) {
            step_state(s, xs[q], ps[q], g1);
            os[q] = 0.5f * (xs[q] + s[7]);
        }
        *(float4*)(op + n) = make_float4(os[0], os[1], os[2], os[3]);
    }
}

// ---------------------------------------------------------------------------
extern "C" void kernel_launch(void* const* d_in, const int* in_sizes, int n_in,
                              void* d_out, int out_size, void* d_ws, size_t ws_size,
                              hipStream_t stream)
{
    const float* x        = (const float*)d_in[0];   // (2, N)
    const float* rate01   = (const float*)d_in[1];
    const float* phoff01  = (const float*)d_in[2];
    const float* W1       = (const float*)d_in[3];   // (1,16)
    const float* b1       = (const float*)d_in[4];   // (16,)
    const float* W2       = (const float*)d_in[5];   // (16,1)
    const float* b2       = (const float*)d_in[6];   // (1,)
    const float* lfo_amp  = (const float*)d_in[7];
    const float* bias     = (const float*)d_in[8];
    const float* depth    = (const float*)d_in[9];
    const float* g1       = (const float*)d_in[10];
    const float* prevph   = (const float*)d_in[11];
    float* out            = (float*)d_out;           // (2, N)

    const int n_per_ch = in_sizes[0] / 2;            // 524288

    // Workspace layout (floats): p[2*N] | T[2*NCHUNKS*72] | S[2*NCHUNKS*8]
    float* wsf   = (float*)d_ws;
    float* p_buf = wsf;
    float* T_buf = p_buf + (size_t)2 * n_per_ch;
    float* S_buf = T_buf + (size_t)2 * NCHUNKS * 72;

    // Stage 1: WMMA coefficient computation (65536 tiles of 16 samples).
    phaser_coeff_wmma<<<256, 256, 0, stream>>>(
        rate01, phoff01, W1, b1, W2, b2, lfo_amp, bias, depth, prevph,
        p_buf, n_per_ch);

    // Stage 2A: per-chunk transfer composition (2048 independent chunks).
    chunk_transfer<<<(2 * NCHUNKS + 255) / 256, 256, 0, stream>>>(
        x, p_buf, g1, T_buf, n_per_ch);

    // Stage 2B: short serial scan (one thread per channel).
    chunk_scan<<<1, 32, 0, stream>>>(T_buf, S_buf);

    // Stage 2C: parallel replay + emit.
    chunk_emit<<<(2 * NCHUNKS + 255) / 256, 256, 0, stream>>>(
        x, p_buf, S_buf, g1, out, n_per_ch);
}